// MessageBuildingLayerLSH_45062796869962
// MI455X (gfx1250) — compile-verified
//
#include <hip/hip_runtime.h>
#include <hip/hip_bf16.h>

// ---------------------------------------------------------------------------
// MessageBuildingLayerLSH for MI455X (gfx1250, wave32, WMMA + async-to-LDS)
//
// Stage 1: LSH projection (204800x128 @ 128x100) with V_WMMA_F32_16X16X4_F32,
//          async global->LDS staging, fused argmax over [mul, -mul] -> bin
//          index, fused per-chunk histograms.
// Stage 2: deterministic chunked counting sort (stable within chunks).
// Stage 3: gather of x_node / msk into binned order (pure bandwidth).
// Stage 4: 1600x (128x128x128) fp32 Gram matrices via WMMA, async-to-LDS
//          staging, fused sqrt/exp/mask epilogue producing dm.
//
// Note: staging skips the x*msk multiply. Since msk is {0,1} and dm is
// scaled by m_i*m_j at the end, unmasked Gram + epilogue masking is exact.
// ---------------------------------------------------------------------------

typedef __attribute__((ext_vector_type(2))) float v2f;
typedef __attribute__((ext_vector_type(8))) float v8f;

#define B_DIM     8
#define N_PTS     25600
#define D_MSG     128
#define D_NODE    256
#define BIN_SIZE  128
#define N_BINS    200            // N_PTS / BIN_SIZE
#define CB_COLS   100            // codebook_slice
#define CB_COLS_P 112            // padded to 7 x 16
#define NBV       400            // bin-value range is [0, 398]; round to 400
#define CHUNKS_PB 200            // N_PTS / 128 chunks per batch
#define N_CHUNKS  (B_DIM * CHUNKS_PB)   // 1600
#define XPAD      132            // LDS row stride (floats); 528 B, 16B-aligned

// Output layout (elements, in reference return order)
#define OUT_BINS_OFF  0
#define OUT_BINS_LEN  (B_DIM * N_PTS)                 // 204800 (int32 bits)
#define OUT_FEAT_OFF  (OUT_BINS_OFF + OUT_BINS_LEN)
#define OUT_FEAT_LEN  (B_DIM * N_PTS * D_NODE)        // 52428800
#define OUT_DM_OFF    (OUT_FEAT_OFF + OUT_FEAT_LEN)
#define OUT_DM_LEN    (B_DIM * N_PTS * BIN_SIZE)      // 26214400
#define OUT_MSK_OFF   (OUT_DM_OFF + OUT_DM_LEN)

// Async copy of 16 bytes global -> LDS (ASYNCcnt-tracked, no VGPR round-trip)
__device__ __forceinline__ void async_copy_b128(const void* gptr, void* lptr) {
  unsigned long long ga = (unsigned long long)(uintptr_t)gptr;
  unsigned int lo = (unsigned int)(uintptr_t)lptr;  // generic LDS addr[31:0] = LDS offset
  asm volatile("global_load_async_to_lds_b128 %0, %1, off"
               :: "v"(lo), "v"(ga) : "memory");
}
__device__ __forceinline__ void async_wait0() {
  asm volatile("s_wait_asynccnt 0x0" ::: "memory");
}

// ---------------------------------------------------------------------------
// Kernel 1: projection + argmax + bin index + per-chunk histogram
// grid = 1600 blocks (one per 128-row chunk), 256 threads (8 waves)
// ---------------------------------------------------------------------------
__global__ __launch_bounds__(256)
void lsh_proj_bin_kernel(const float* __restrict__ x_msg,
                         const float* __restrict__ msk,
                         const float* __restrict__ codebook,
                         int* __restrict__ bin_idx,
                         int* __restrict__ counts) {
  __shared__ float sCB[D_MSG * CB_COLS_P];              // 57344 B, zero-padded
  __shared__ alignas(16) float sX[128 * XPAD];          // 67584 B, X then C
  __shared__ int   sHist[NBV];

  const int tid  = threadIdx.x;
  const int base = blockIdx.x * 128;         // flattened row base over B*N

  // Kick off async staging of 128 rows of x_msg into padded LDS rows
  {
    const float* src = x_msg + (size_t)base * D_MSG;
    for (int i = tid; i < 128 * (D_MSG / 4); i += 256) {
      int r = i / (D_MSG / 4), c4 = i % (D_MSG / 4);
      async_copy_b128(src + (size_t)r * D_MSG + c4 * 4, &sX[r * XPAD + c4 * 4]);
    }
  }

  for (int i = tid; i < NBV; i += 256) sHist[i] = 0;

  // Stage codebook (128 x 100, row-major) into padded LDS (overlaps async)
  for (int i = tid; i < D_MSG * CB_COLS_P; i += 256) {
    int k = i / CB_COLS_P, n = i % CB_COLS_P;
    sCB[i] = (n < CB_COLS) ? codebook[k * CB_COLS + n] : 0.0f;
  }
  async_wait0();
  __syncthreads();

  const int wave = tid >> 5, lane = tid & 31;
  const int hi = lane >> 4, lm = lane & 15;
  const float* xrow = &sX[(wave * 16 + lm) * XPAD];

  v8f acc[7];
#pragma unroll
  for (int t = 0; t < 7; ++t) acc[t] = (v8f){0.f,0.f,0.f,0.f,0.f,0.f,0.f,0.f};

#pragma unroll 4
  for (int kb = 0; kb < D_MSG; kb += 4) {
    v2f a;
    a.x = xrow[kb + 2 * hi];
    a.y = xrow[kb + 2 * hi + 1];
#pragma unroll
    for (int ct = 0; ct < 7; ++ct) {
      v2f bm;
      bm.x = sCB[(kb + 2 * hi) * CB_COLS_P + ct * 16 + lm];
      bm.y = sCB[(kb + 2 * hi + 1) * CB_COLS_P + ct * 16 + lm];
      acc[ct] = __builtin_amdgcn_wmma_f32_16x16x4_f32(
          false, a, false, bm, (short)0, acc[ct], false, false);
    }
  }

  // Write C tiles back into this wave's own 16 rows of sX (aliasing is safe:
  // each wave only ever reads/writes its own rows, LDS is in-order per wave).
#pragma unroll
  for (int ct = 0; ct < 7; ++ct)
#pragma unroll
    for (int v = 0; v < 8; ++v)
      sX[(wave * 16 + v + 8 * hi) * XPAD + ct * 16 + lm] = acc[ct][v];

  // argmax over cmul = [mul, -mul]: lanes 0..15 of each wave own one row each
  if (hi == 0) {
    const float* row = &sX[(wave * 16 + lm) * XPAD];
    float mx = row[0]; int imax = 0;
    float mn = row[0]; int imin = 0;
    for (int j = 1; j < CB_COLS; ++j) {
      float v = row[j];
      if (v > mx) { mx = v; imax = j; }
      if (v < mn) { mn = v; imin = j; }
    }
    int a = (mx >= -mn) ? imax : (CB_COLS + imin);   // first-occurrence ties
    int g = base + wave * 16 + lm;
    float m = msk[g];
    int bv = a + ((m != 0.0f) ? 0 : (N_BINS - 1));
    bin_idx[g] = bv;
    atomicAdd(&sHist[bv], 1);
  }
  __syncthreads();
  for (int i = tid; i < NBV; i += 256)
    counts[(size_t)blockIdx.x * NBV + i] = sHist[i];
}

// ---------------------------------------------------------------------------
// Kernel 2: per-batch (bin x chunk) exclusive offsets. grid = 8, 512 threads.
// ---------------------------------------------------------------------------
__global__ __launch_bounds__(512)
void sort_offsets_kernel(const int* __restrict__ counts,
                         int* __restrict__ offs) {
  __shared__ int sTot[NBV];
  __shared__ int sBase[NBV];
  const int b = blockIdx.x, t = threadIdx.x;
  if (t < NBV) {
    int s = 0;
    for (int c = 0; c < CHUNKS_PB; ++c)
      s += counts[(size_t)(b * CHUNKS_PB + c) * NBV + t];
    sTot[t] = s;
  }
  __syncthreads();
  if (t == 0) {
    int s = 0;
    for (int i = 0; i < NBV; ++i) { sBase[i] = s; s += sTot[i]; }
  }
  __syncthreads();
  if (t < NBV) {
    int run = sBase[t];
    for (int c = 0; c < CHUNKS_PB; ++c) {
      size_t o = (size_t)(b * CHUNKS_PB + c) * NBV + t;
      offs[o] = run;
      run += counts[o];
    }
  }
}

// ---------------------------------------------------------------------------
// Kernel 3: stable placement per chunk. grid = 1600, 128 threads.
// ---------------------------------------------------------------------------
__global__ __launch_bounds__(128)
void sort_place_kernel(const int* __restrict__ bin_idx,
                       const int* __restrict__ offs,
                       int* __restrict__ bins_out) {
  __shared__ int sOffs[NBV];
  __shared__ int sBin[128];
  const int chunk = blockIdx.x;
  const int b = chunk / CHUNKS_PB;
  const int ci = chunk % CHUNKS_PB;
  const int t = threadIdx.x;
  for (int i = t; i < NBV; i += 128) sOffs[i] = offs[(size_t)chunk * NBV + i];
  sBin[t] = bin_idx[(size_t)chunk * 128 + t];
  __syncthreads();
  if (t == 0) {                       // serial within chunk => stable
    int* outb = bins_out + (size_t)b * N_PTS;
    for (int i = 0; i < 128; ++i) {
      int pos = sOffs[sBin[i]]++;
      outb[pos] = ci * 128 + i;
    }
  }
}

// ---------------------------------------------------------------------------
// Kernel 4: gather x_node + msk into binned order. grid = B*N, 256 threads.
// ---------------------------------------------------------------------------
__global__ __launch_bounds__(256)
void gather_kernel(const float* __restrict__ x_node,
                   const float* __restrict__ msk,
                   const int* __restrict__ bins,
                   float* __restrict__ out_feat,
                   float* __restrict__ out_msk) {
  const int p = blockIdx.x;           // binned flat row
  const int b = p / N_PTS;
  const int idx = bins[p];
  const float* src = x_node + ((size_t)b * N_PTS + idx) * D_NODE;
  out_feat[(size_t)p * D_NODE + threadIdx.x] = src[threadIdx.x];
  if (threadIdx.x == 0) out_msk[p] = msk[b * N_PTS + idx];
}

// ---------------------------------------------------------------------------
// Kernel 5: per-bin Gram matrix via WMMA + distance epilogue.
// grid = B*N_BINS = 1600, 256 threads (8 waves, 8 tiles each).
// ---------------------------------------------------------------------------
__global__ __launch_bounds__(256)
void pairwise_kernel(const float* __restrict__ x_msg,
                     const float* __restrict__ msk,
                     const int* __restrict__ bins,
                     float* __restrict__ dm) {
  __shared__ alignas(16) float sX[BIN_SIZE * XPAD];   // gathered msg tile
  __shared__ float sNorm[BIN_SIZE];
  __shared__ float sM[BIN_SIZE];

  const int blk = blockIdx.x;             // b * N_BINS + bin
  const int b = blk / N_BINS;
  const int tid = threadIdx.x;

  // Gather 128 rows of x_msg via async-to-LDS (2 threads per row, B128)
  {
    int r = tid >> 1, half = tid & 1;
    int idx = bins[(size_t)blk * BIN_SIZE + r];
    const float* srcrow = x_msg + ((size_t)b * N_PTS + idx) * D_MSG;
    for (int c4 = half * 16; c4 < half * 16 + 16; ++c4)
      async_copy_b128(srcrow + c4 * 4, &sX[r * XPAD + c4 * 4]);
    if (half == 0) sM[r] = msk[(size_t)b * N_PTS + idx];
  }
  async_wait0();
  __syncthreads();
  if (tid < BIN_SIZE) {                   // unmasked norms (exact, see header)
    const float* row = &sX[tid * XPAD];
    float s = 0.f;
    for (int c = 0; c < D_MSG; ++c) s += row[c] * row[c];
    sNorm[tid] = s;
  }
  __syncthreads();

  const int wave = tid >> 5, lane = tid & 31;
  const int hi = lane >> 4, lm = lane & 15;
  float* out = dm + (size_t)blk * BIN_SIZE * BIN_SIZE;

  for (int t8 = 0; t8 < 8; ++t8) {
    int t = wave * 8 + t8;
    int ti = t >> 3, tj = t & 7;
    const float* arow = &sX[(ti * 16 + lm) * XPAD];
    const float* brow = &sX[(tj * 16 + lm) * XPAD];
    v8f acc = (v8f){0.f,0.f,0.f,0.f,0.f,0.f,0.f,0.f};
#pragma unroll 8
    for (int kb = 0; kb < D_MSG; kb += 4) {
      v2f a, bm;
      a.x  = arow[kb + 2 * hi]; a.y  = arow[kb + 2 * hi + 1];
      bm.x = brow[kb + 2 * hi]; bm.y = brow[kb + 2 * hi + 1];
      acc = __builtin_amdgcn_wmma_f32_16x16x4_f32(
          false, a, false, bm, (short)0, acc, false, false);
    }
#pragma unroll
    for (int v = 0; v < 8; ++v) {
      int gi = ti * 16 + v + 8 * hi;
      int gj = tj * 16 + lm;
      float mm = sM[gi] * sM[gj];
      float d2 = sNorm[gi] + sNorm[gj] - 2.0f * acc[v];
      d2 = fminf(fmaxf(d2, 1e-6f), 1e6f);
      float val = __expf(-0.1f * sqrtf(d2));
      val = fminf(fmaxf(val, 0.0f), 1.0f);
      out[(size_t)gi * BIN_SIZE + gj] = val * mm;
    }
  }
}

// ---------------------------------------------------------------------------
extern "C" void kernel_launch(void* const* d_in, const int* in_sizes, int n_in,
                              void* d_out, int out_size, void* d_ws, size_t ws_size,
                              hipStream_t stream) {
  const float* x_msg    = (const float*)d_in[0];
  const float* x_node   = (const float*)d_in[1];
  const float* msk      = (const float*)d_in[2];
  const float* codebook = (const float*)d_in[3];

  float* out       = (float*)d_out;
  int*   bins_out  = (int*)(out + OUT_BINS_OFF);   // int32 bits in-place
  float* out_feat  = out + OUT_FEAT_OFF;
  float* out_dm    = out + OUT_DM_OFF;
  float* out_mskb  = out + OUT_MSK_OFF;

  int* ws_binidx = (int*)d_ws;                       // B*N ints
  int* ws_counts = ws_binidx + B_DIM * N_PTS;        // N_CHUNKS*NBV ints
  int* ws_offs   = ws_counts + N_CHUNKS * NBV;       // N_CHUNKS*NBV ints

  lsh_proj_bin_kernel<<<N_CHUNKS, 256, 0, stream>>>(x_msg, msk, codebook,
                                                    ws_binidx, ws_counts);
  sort_offsets_kernel<<<B_DIM, 512, 0, stream>>>(ws_counts, ws_offs);
  sort_place_kernel<<<N_CHUNKS, 128, 0, stream>>>(ws_binidx, ws_offs, bins_out);
  gather_kernel<<<B_DIM * N_PTS, 256, 0, stream>>>(x_node, msk, bins_out,
                                                   out_feat, out_mskb);
  pairwise_kernel<<<B_DIM * N_BINS, 256, 0, stream>>>(x_msg, msk, bins_out,
                                                      out_dm);
}